// SVF_3D_12103217840716
// MI455X (gfx1250) — compile-verified
//
#include <hip/hip_runtime.h>
#include <stdint.h>

// Geometry: N=2, C=3, D=H=W=128
static constexpr int  CHANNEL  = 128 * 128 * 128;  // 2^21 elements per channel
static constexpr int  NSTR     = 3 * CHANNEL;      // per-batch stride (elements)
static constexpr long NE       = 2L * NSTR;        // one output tensor (12,582,912 floats)
static constexpr int  NO_STEPS = 32;

// ---------------------------------------------------------------------------
// One scaling-and-squaring step:
//   dst = s*src + trilinear_sample(s*src, identity + s*src)
// with s = 2^-32 on the first step (bit-exact fusion of the init pass, since
// power-of-two scaling distributes exactly over fp32 add/mul) and s = 1
// otherwise.  Border clamp + align_corners=true, mirroring the reference's
// arithmetic order exactly.
//
// The block's own 3-channel displacement values are staged into LDS with the
// CDNA5 async global->LDS DMA path (ASYNCcnt); the gathers are 12 paired
// (x0,x0+1) 8-byte loads with the x-border folded into the weights once.
// ---------------------------------------------------------------------------
template <bool FIRST>
__global__ __launch_bounds__(256) void svf_step(const float* __restrict__ src,
                                                float* __restrict__ dst) {
    __shared__ float s_self[3 * 256];

    const int tid      = threadIdx.x;
    const int gidBlock = blockIdx.x * 256;                  // block-uniform
    const int n        = gidBlock >> 21;                    // CHANNEL = 2^21
    const int lin0     = gidBlock & (CHANNEL - 1);
    const float* base  = src + (long)n * NSTR;

    // ---- async stage: 3 channels x 256 voxels (3 KB) into LDS --------------
    {
        // Low 32 bits of a generic shared pointer are the raw LDS byte address
        uint32_t ldsb  = (uint32_t)(uintptr_t)(&s_self[0]);
        uint64_t saddr = (uint64_t)(uintptr_t)(base + lin0); // uniform -> SGPR pair
        if (tid < 192) {                                     // 192 lanes x 16B = 3 KB
            int c   = tid >> 6;                              // channel 0..2
            int grp = tid & 63;                              // 16B group within channel
            uint32_t voff = (uint32_t)c * (uint32_t)(CHANNEL * 4) + (uint32_t)grp * 16u;
            uint32_t la   = ldsb + (uint32_t)c * 1024u + (uint32_t)grp * 16u;
            asm volatile("global_load_async_to_lds_b128 %0, %1, %2"
                         :: "v"(la), "v"(voff), "s"(saddr)
                         : "memory");
        }
        asm volatile("s_wait_asynccnt 0" ::: "memory");
    }
    __syncthreads();

    const int lin = lin0 + tid;
    const int x   = lin & 127;
    const int y   = (lin >> 7) & 127;
    const int z   = lin >> 14;

    float dx = s_self[tid];
    float dy = s_self[256 + tid];
    float dz = s_self[512 + tid];
    if constexpr (FIRST) {                                  // disp0 = v * 2^-32 (exact)
        dx *= 0x1p-32f; dy *= 0x1p-32f; dz *= 0x1p-32f;
    }

    // transformation = identity + disp ; i = clip((g+d+1)*0.5*(W-1), 0, W-1)
    const float invstep = 2.0f / 127.0f;
    float ix = ((-1.0f + x * invstep) + dx + 1.0f) * 63.5f;
    float iy = ((-1.0f + y * invstep) + dy + 1.0f) * 63.5f;
    float iz = ((-1.0f + z * invstep) + dz + 1.0f) * 63.5f;
    ix = fminf(fmaxf(ix, 0.0f), 127.0f);
    iy = fminf(fmaxf(iy, 0.0f), 127.0f);
    iz = fminf(fmaxf(iz, 0.0f), 127.0f);

    const float x0f = floorf(ix), y0f = floorf(iy), z0f = floorf(iz);
    const float fx = ix - x0f, fy = iy - y0f, fz = iz - z0f;
    const int x0 = (int)x0f, y0 = (int)y0f, z0 = (int)z0f;
    const int y1 = min(y0 + 1, 127);
    const int z1 = min(z0 + 1, 127);

    // x taps as one adjacent pair; fold the x-border (x0==127, fx==0) into the
    // weights: both taps collapse onto element 127 == pair[1] with weight 1.
    const bool  xedge = (x0 == 127);
    const int   xa    = xedge ? 126 : x0;
    const float wxl   = xedge ? 0.0f : (1.0f - fx);
    const float wxh   = xedge ? 1.0f : fx;

    const int o00 = (((z0 << 7) + y0) << 7) + xa;
    const int o01 = (((z0 << 7) + y1) << 7) + xa;
    const int o10 = (((z1 << 7) + y0) << 7) + xa;
    const int o11 = (((z1 << 7) + y1) << 7) + xa;

    const float wy0 = 1.0f - fy, wz0 = 1.0f - fz;
    const float w000 = wz0 * wy0 * wxl, w001 = wz0 * wy0 * wxh;
    const float w010 = wz0 * fy  * wxl, w011 = wz0 * fy  * wxh;
    const float w100 = fz  * wy0 * wxl, w101 = fz  * wy0 * wxh;
    const float w110 = fz  * fy  * wxl, w111 = fz  * fy  * wxh;

    float acc[3] = {dx, dy, dz};
    const float* p = base;
#pragma unroll
    for (int c = 0; c < 3; ++c) {
        float a00[2], a01[2], a10[2], a11[2];
        __builtin_memcpy(a00, p + o00, 8);
        __builtin_memcpy(a01, p + o01, 8);
        __builtin_memcpy(a10, p + o10, 8);
        __builtin_memcpy(a11, p + o11, 8);
        float sval = w000 * a00[0] + w001 * a00[1]
                   + w010 * a01[0] + w011 * a01[1]
                   + w100 * a10[0] + w101 * a10[1]
                   + w110 * a11[0] + w111 * a11[1];
        if constexpr (FIRST) sval *= 0x1p-32f;              // exact pow2 scale
        acc[c] += sval;
        p += CHANNEL;
    }

    float* dbase = dst + (long)n * NSTR + lin;
    dbase[0]           = acc[0];
    dbase[CHANNEL]     = acc[1];
    dbase[2 * CHANNEL] = acc[2];
}

// ---------------------------------------------------------------------------
// transformation = identity(channel-dependent linspace) + displacement
// ---------------------------------------------------------------------------
__global__ __launch_bounds__(256) void svf_final(const float* __restrict__ disp,
                                                 float* __restrict__ trans) {
    long i = (long)blockIdx.x * 256 + threadIdx.x;          // over 2*NSTR elements
    int lin = (int)(i & (CHANNEL - 1));
    int nc  = (int)(i >> 21);                               // n*3 + c, 0..5
    int nn  = nc / 3;
    int c   = nc - 3 * nn;
    int x = lin & 127, y = (lin >> 7) & 127, z = lin >> 14;
    int coord = (c == 0) ? x : ((c == 1) ? y : z);
    float g = -1.0f + coord * (2.0f / 127.0f);
    trans[i] = g + disp[i];
}

// ---------------------------------------------------------------------------
extern "C" void kernel_launch(void* const* d_in, const int* in_sizes, int n_in,
                              void* d_out, int out_size, void* d_ws, size_t ws_size,
                              hipStream_t stream) {
    const float* v = (const float*)d_in[0];                 // [2,3,128,128,128]
    float* out   = (float*)d_out;
    float* trans = out;        // first NE floats  (transformation output)
    float* disp  = out + NE;   // second NE floats (displacement output)

    const unsigned stepBlocks = (unsigned)((2L * CHANNEL) / 256);   // 16384

    // Step 0 fused with init: reads v, scales by 2^-32 (bit-exact), -> trans.
    svf_step<true><<<stepBlocks, 256, 0, stream>>>(v, trans);

    // Steps 1..31 ping-pong; odd steps land in `disp`, so step 31 ends there.
    for (int s = 1; s < NO_STEPS; ++s) {
        float* sPtr = (s & 1) ? trans : disp;
        float* dPtr = (s & 1) ? disp  : trans;
        svf_step<false><<<stepBlocks, 256, 0, stream>>>(sPtr, dPtr);
    }

    // transformation = identity + displacement -> transformation region
    svf_final<<<(unsigned)((2L * NSTR) / 256), 256, 0, stream>>>(disp, trans);
}